// GroupedExpertMLP_69234872811781
// MI455X (gfx1250) — compile-verified
//
#include <hip/hip_runtime.h>

// MoE grouped SwiGLU MLP for MI455X (gfx1250).
//
// Roofline: 12.9 GFLOP vs 805 MB of single-use fp32 weights -> memory bound
// (~35us floor @ 23.3 TB/s). So we keep full fp32 precision and use the
// fp32 WMMA path (v_wmma_f32_16x16x4_f32): zero cvt overhead, the matrix
// pipe is nowhere near the bottleneck.
//
// Fragment feeding: for 16x16x4 f32 WMMA, each lane holds 2 f32 of A and B,
// slot = 2*(lane>>4) + vgpr. One contiguous b128 per lane at element offset
// kb + 4*(lane>>4) yields TWO valid K-partitions ({kb,kb+1,kb+4,kb+5} and
// {kb+2,kb+3,kb+6,kb+7}); A and B use the identical mapping so the GEMM sum
// is exact. Inner loop: 4 global_load_b128 : 8 v_wmma per K=8.
//
// Weights are streamed non-temporally (single use, 805MB >> 192MB L2) so L2
// retains x / hidden, which are re-read ~64x per expert.

typedef __attribute__((ext_vector_type(2))) float v2f;
typedef __attribute__((ext_vector_type(4))) float f32x4;
typedef __attribute__((ext_vector_type(8))) float v8f;

#define MOE_D 1024  // d_model
#define MOE_F 1024  // d_ff

__device__ __forceinline__ v8f wmma4f32(v2f a, v2f b, v8f c) {
  // (neg_a, A, neg_b, B, c_mod, C, reuse_a, reuse_b)
  return __builtin_amdgcn_wmma_f32_16x16x4_f32(false, a, false, b, (short)0, c,
                                               false, false);
}

// ---------------------------------------------------------------------------
// Kernel 1: gate/up projections + SwiGLU -> hidden (fp32) in workspace.
// grid = (F/128, n_segments), block = 256 (8 waves). Wave w owns f-tile
// [ftile, ftile+16) and both 16-row token tiles of its 32-token segment.
// ---------------------------------------------------------------------------
__global__ __launch_bounds__(256)
void moe_gateup_kernel(const float* __restrict__ x,
                       const float* __restrict__ w1,
                       const float* __restrict__ w3,
                       const int* __restrict__ expert_ids,
                       const int* __restrict__ seg_starts,
                       float* __restrict__ hidden) {
  const int s    = blockIdx.y;
  const int e    = expert_ids[s];
  const int row0 = seg_starts[s];

  const int wave = threadIdx.x >> 5;
  const int lane = threadIdx.x & 31;
  const int ln   = lane & 15;   // N / M index within tile
  const int lh   = lane >> 4;   // lane-half selects K slots

  const int ftile = blockIdx.x * 128 + wave * 16;

  // A: token rows (contiguous in d). B: W rows f (contiguous in d).
  const float* a0 = x + (size_t)(row0 + ln) * MOE_D + 4 * lh;
  const float* a1 = a0 + (size_t)16 * MOE_D;
  const float* b1 = w1 + ((size_t)e * MOE_F + (ftile + ln)) * MOE_D + 4 * lh;
  const float* b3 = w3 + ((size_t)e * MOE_F + (ftile + ln)) * MOE_D + 4 * lh;

  v8f g0 = {}, g1 = {}, u0 = {}, u1 = {};

#pragma unroll 4
  for (int kb = 0; kb < MOE_D; kb += 8) {
    f32x4 qa0 = *(const f32x4*)(a0 + kb);
    f32x4 qa1 = *(const f32x4*)(a1 + kb);
    f32x4 qb1 = __builtin_nontemporal_load((const f32x4*)(b1 + kb));
    f32x4 qb3 = __builtin_nontemporal_load((const f32x4*)(b3 + kb));

    v2f aP0 = {qa0.x, qa0.y}, aQ0 = {qa0.z, qa0.w};
    v2f aP1 = {qa1.x, qa1.y}, aQ1 = {qa1.z, qa1.w};
    v2f bP1 = {qb1.x, qb1.y}, bQ1 = {qb1.z, qb1.w};
    v2f bP3 = {qb3.x, qb3.y}, bQ3 = {qb3.z, qb3.w};

    g0 = wmma4f32(aP0, bP1, g0);
    g0 = wmma4f32(aQ0, bQ1, g0);
    g1 = wmma4f32(aP1, bP1, g1);
    g1 = wmma4f32(aQ1, bQ1, g1);
    u0 = wmma4f32(aP0, bP3, u0);
    u0 = wmma4f32(aQ0, bQ3, u0);
    u1 = wmma4f32(aP1, bP3, u1);
    u1 = wmma4f32(aQ1, bQ3, u1);
  }

  // C/D layout: VGPR i -> row m = i + 8*lh, col n = ln.
  const int fcol = ftile + ln;
#pragma unroll
  for (int i = 0; i < 8; ++i) {
    const int m = i + 8 * lh;
    float ga = g0[i], ua = u0[i];
    float h  = (ga / (1.0f + __expf(-ga))) * ua;  // silu(g) * u
    hidden[(size_t)(row0 + m) * MOE_F + fcol] = h;
    ga = g1[i];
    ua = u1[i];
    h  = (ga / (1.0f + __expf(-ga))) * ua;
    hidden[(size_t)(row0 + 16 + m) * MOE_F + fcol] = h;
  }
}

// ---------------------------------------------------------------------------
// Kernel 2: down projection hidden (32 x F) x W2(D,F)^T -> out (32 x D).
// grid = (D/128, n_segments), block = 256. Same streaming structure.
// ---------------------------------------------------------------------------
__global__ __launch_bounds__(256)
void moe_down_kernel(const float* __restrict__ hidden,
                     const float* __restrict__ w2,
                     const int* __restrict__ expert_ids,
                     const int* __restrict__ seg_starts,
                     float* __restrict__ out) {
  const int s    = blockIdx.y;
  const int e    = expert_ids[s];
  const int row0 = seg_starts[s];

  const int wave = threadIdx.x >> 5;
  const int lane = threadIdx.x & 31;
  const int ln   = lane & 15;
  const int lh   = lane >> 4;

  const int dtile = blockIdx.x * 128 + wave * 16;

  const float* a0 = hidden + (size_t)(row0 + ln) * MOE_F + 4 * lh;
  const float* a1 = a0 + (size_t)16 * MOE_F;
  const float* b2 = w2 + ((size_t)e * MOE_D + (dtile + ln)) * MOE_F + 4 * lh;

  v8f c0 = {}, c1 = {};

#pragma unroll 4
  for (int kb = 0; kb < MOE_F; kb += 8) {
    f32x4 qa0 = *(const f32x4*)(a0 + kb);
    f32x4 qa1 = *(const f32x4*)(a1 + kb);
    f32x4 qb2 = __builtin_nontemporal_load((const f32x4*)(b2 + kb));

    v2f aP0 = {qa0.x, qa0.y}, aQ0 = {qa0.z, qa0.w};
    v2f aP1 = {qa1.x, qa1.y}, aQ1 = {qa1.z, qa1.w};
    v2f bP  = {qb2.x, qb2.y}, bQ  = {qb2.z, qb2.w};

    c0 = wmma4f32(aP0, bP, c0);
    c0 = wmma4f32(aQ0, bQ, c0);
    c1 = wmma4f32(aP1, bP, c1);
    c1 = wmma4f32(aQ1, bQ, c1);
  }

  const int dcol = dtile + ln;
#pragma unroll
  for (int i = 0; i < 8; ++i) {
    const int m = i + 8 * lh;
    __builtin_nontemporal_store(c0[i],
        out + (size_t)(row0 + m) * MOE_D + dcol);
    __builtin_nontemporal_store(c1[i],
        out + (size_t)(row0 + 16 + m) * MOE_D + dcol);
  }
}

// ---------------------------------------------------------------------------
extern "C" void kernel_launch(void* const* d_in, const int* in_sizes, int n_in,
                              void* d_out, int out_size, void* d_ws,
                              size_t ws_size, hipStream_t stream) {
  const float* x   = (const float*)d_in[0];
  const float* w1  = (const float*)d_in[1];
  const float* w3  = (const float*)d_in[2];
  const float* w2  = (const float*)d_in[3];
  const int*   eid = (const int*)d_in[4];
  const int*   sst = (const int*)d_in[5];

  const int n_seg = in_sizes[4];         // number of expert segments (64)
  float* hidden   = (float*)d_ws;        // T x F fp32 (8 MB)

  dim3 block(256);
  dim3 grid1(MOE_F / 128, n_seg);
  moe_gateup_kernel<<<grid1, block, 0, stream>>>(x, w1, w3, eid, sst, hidden);

  dim3 grid2(MOE_D / 128, n_seg);
  moe_down_kernel<<<grid2, block, 0, stream>>>(hidden, w2, eid, sst,
                                               (float*)d_out);
}